// PredictionModel_1846835938207
// MI455X (gfx1250) — compile-verified
//
#include <hip/hip_runtime.h>
#include <hip/hip_bf16.h>

typedef __attribute__((ext_vector_type(2))) float v2f;
typedef __attribute__((ext_vector_type(8))) float v8f;

#define BB 4
#define LL 512
#define DD 512
#define CC 256
#define KK 12
#define LW 500          // L - K
#define MTOT (BB * LW)  // 2000 = 125 * 16
#define NTOT (KK * DD)  // 6144

// ---------------------------------------------------------------------------
// Kernel A: c_proj[m, j] = sum_c A[m,c] * W[j,c]
//   A[m,c] = c[(b*LL + l)*CC + c], m = b*LW + l
//   W[j,c] = Wk[j*CC + c],         j = k*DD + d
// One wave computes one 16-row M-tile x four 16-col N-tiles (A reuse 4x).
// Block = 256 threads = 8 waves -> 8*64 = 512 N columns per block.
// Grid  = (125 M-tiles, 12 N-groups).
// f32 WMMA 16x16x4: A/B fragment = v2f per lane, 64 steps over C=256.
// ---------------------------------------------------------------------------
__global__ __launch_bounds__(256) void cproj_gemm_kernel(
    const float* __restrict__ c, const float* __restrict__ Wk,
    float* __restrict__ cproj) {
  const unsigned lane = threadIdx.x & 31u;
  const unsigned wave = threadIdx.x >> 5;

  const unsigned mtile = blockIdx.x;                 // 0..124
  const unsigned nbase = blockIdx.y * 512u + wave * 64u;  // this wave's first col

  // A-fragment addressing: row = mtile*16 + lane%16, K-pair offset = (lane/16)*2
  const unsigned mrow = mtile * 16u + (lane & 15u);
  const unsigned b_idx = mrow / LW;
  const unsigned l_idx = mrow % LW;
  const float* arow = c + ((size_t)(b_idx * LL + l_idx)) * CC;
  const unsigned koff = (lane >> 4) * 2u;

  // B-fragment rows: j_t = nbase + t*16 + lane%16
  const float* brow0 = Wk + (size_t)(nbase + 0u * 16u + (lane & 15u)) * CC;
  const float* brow1 = Wk + (size_t)(nbase + 1u * 16u + (lane & 15u)) * CC;
  const float* brow2 = Wk + (size_t)(nbase + 2u * 16u + (lane & 15u)) * CC;
  const float* brow3 = Wk + (size_t)(nbase + 3u * 16u + (lane & 15u)) * CC;

  v8f acc0 = {}, acc1 = {}, acc2 = {}, acc3 = {};

#pragma unroll 4
  for (unsigned cs = 0; cs < CC; cs += 4) {
    const v2f a  = *(const v2f*)(arow  + cs + koff);
    const v2f b0 = *(const v2f*)(brow0 + cs + koff);
    const v2f b1 = *(const v2f*)(brow1 + cs + koff);
    const v2f b2 = *(const v2f*)(brow2 + cs + koff);
    const v2f b3 = *(const v2f*)(brow3 + cs + koff);
    acc0 = __builtin_amdgcn_wmma_f32_16x16x4_f32(false, a, false, b0, (short)0, acc0, false, false);
    acc1 = __builtin_amdgcn_wmma_f32_16x16x4_f32(false, a, false, b1, (short)0, acc1, false, false);
    acc2 = __builtin_amdgcn_wmma_f32_16x16x4_f32(false, a, false, b2, (short)0, acc2, false, false);
    acc3 = __builtin_amdgcn_wmma_f32_16x16x4_f32(false, a, false, b3, (short)0, acc3, false, false);
  }

  // D layout: VGPR i -> row = i + 8*(lane/16), col = lane%16
  const unsigned rbase = mtile * 16u + (lane >> 4) * 8u;
  const unsigned cidx = lane & 15u;
#pragma unroll
  for (unsigned i = 0; i < 8; ++i) {
    const size_t row = (size_t)(rbase + i) * NTOT;
    cproj[row + nbase + 0u * 16u + cidx] = acc0[i];
    cproj[row + nbase + 1u * 16u + cidx] = acc1[i];
    cproj[row + nbase + 2u * 16u + cidx] = acc2[i];
    cproj[row + nbase + 3u * 16u + cidx] = acc3[i];
  }
}

// ---------------------------------------------------------------------------
// Kernel B: nine dot products per (b, l, k), one wave each.
//   out[b, 0,   l, k] = <cproj[b,l,k,:], z[b, l+1+k, :]>
//   out[b, 1+n, l, k] = <cproj[b,l,k,:], z[perm_B[n], perm_L[l]+1+k, :]>
//   out[b, 5+n, l, k] = <cproj[b,l,k,:], z[b, perms_len[n,l]+1+k, :]>
// ---------------------------------------------------------------------------
__global__ __launch_bounds__(256) void dots_kernel(
    const float* __restrict__ cproj, const float* __restrict__ z,
    const int* __restrict__ perms_len, const int* __restrict__ perm_L,
    const int* __restrict__ perm_B, float* __restrict__ out) {
  const unsigned lane = threadIdx.x & 31u;
  const unsigned wave = threadIdx.x >> 5;
  const unsigned wg = blockIdx.x * 8u + wave;  // 0..23999
  if (wg >= (unsigned)(BB * LW * KK)) return;

  const unsigned k = wg % KK;
  const unsigned l = (wg / KK) % LW;
  const unsigned b = wg / (KK * LW);

  const float* cp = cproj + (size_t)(b * LW + l) * NTOT + (size_t)k * DD;

  const float* z0 = z + (size_t)(b * LL + (l + 1u + k)) * DD;

  const unsigned pl = (unsigned)perm_L[l];
  const float* zg0 = z + (size_t)((unsigned)perm_B[0] * LL + (pl + 1u + k)) * DD;
  const float* zg1 = z + (size_t)((unsigned)perm_B[1] * LL + (pl + 1u + k)) * DD;
  const float* zg2 = z + (size_t)((unsigned)perm_B[2] * LL + (pl + 1u + k)) * DD;
  const float* zg3 = z + (size_t)((unsigned)perm_B[3] * LL + (pl + 1u + k)) * DD;

  const float* zl0 = z + (size_t)(b * LL + ((unsigned)perms_len[0 * LW + l] + 1u + k)) * DD;
  const float* zl1 = z + (size_t)(b * LL + ((unsigned)perms_len[1 * LW + l] + 1u + k)) * DD;
  const float* zl2 = z + (size_t)(b * LL + ((unsigned)perms_len[2 * LW + l] + 1u + k)) * DD;
  const float* zl3 = z + (size_t)(b * LL + ((unsigned)perms_len[3 * LW + l] + 1u + k)) * DD;

  float s0 = 0.f;
  float sg0 = 0.f, sg1 = 0.f, sg2 = 0.f, sg3 = 0.f;
  float sl0 = 0.f, sl1 = 0.f, sl2 = 0.f, sl3 = 0.f;

#pragma unroll 4
  for (unsigned it = 0; it < DD / 32u; ++it) {
    const unsigned d = lane + it * 32u;
    const float cv = cp[d];
    s0  = fmaf(cv, z0[d],  s0);
    sg0 = fmaf(cv, zg0[d], sg0);
    sg1 = fmaf(cv, zg1[d], sg1);
    sg2 = fmaf(cv, zg2[d], sg2);
    sg3 = fmaf(cv, zg3[d], sg3);
    sl0 = fmaf(cv, zl0[d], sl0);
    sl1 = fmaf(cv, zl1[d], sl1);
    sl2 = fmaf(cv, zl2[d], sl2);
    sl3 = fmaf(cv, zl3[d], sl3);
  }

  // wave32 butterfly reduction
#pragma unroll
  for (int off = 16; off > 0; off >>= 1) {
    s0  += __shfl_xor(s0,  off, 32);
    sg0 += __shfl_xor(sg0, off, 32);
    sg1 += __shfl_xor(sg1, off, 32);
    sg2 += __shfl_xor(sg2, off, 32);
    sg3 += __shfl_xor(sg3, off, 32);
    sl0 += __shfl_xor(sl0, off, 32);
    sl1 += __shfl_xor(sl1, off, 32);
    sl2 += __shfl_xor(sl2, off, 32);
    sl3 += __shfl_xor(sl3, off, 32);
  }

  if (lane == 0) {
    // out shape (B, 1+B+B, LW, K)
    const size_t base = (size_t)b * (1 + BB + BB) * LW * KK;
    const size_t lk = (size_t)l * KK + k;
    out[base + 0 * (size_t)(LW * KK) + lk] = s0;
    out[base + 1 * (size_t)(LW * KK) + lk] = sg0;
    out[base + 2 * (size_t)(LW * KK) + lk] = sg1;
    out[base + 3 * (size_t)(LW * KK) + lk] = sg2;
    out[base + 4 * (size_t)(LW * KK) + lk] = sg3;
    out[base + 5 * (size_t)(LW * KK) + lk] = sl0;
    out[base + 6 * (size_t)(LW * KK) + lk] = sl1;
    out[base + 7 * (size_t)(LW * KK) + lk] = sl2;
    out[base + 8 * (size_t)(LW * KK) + lk] = sl3;
  }
}

extern "C" void kernel_launch(void* const* d_in, const int* in_sizes, int n_in,
                              void* d_out, int out_size, void* d_ws, size_t ws_size,
                              hipStream_t stream) {
  const float* c = (const float*)d_in[0];
  const float* z = (const float*)d_in[1];
  const float* Wk = (const float*)d_in[2];
  const int* perms_len = (const int*)d_in[3];
  const int* perm_L = (const int*)d_in[4];
  const int* perm_B = (const int*)d_in[5];
  float* out = (float*)d_out;
  float* cproj = (float*)d_ws;  // needs MTOT*NTOT*4 = 49,152,000 bytes

  dim3 gridA(MTOT / 16, NTOT / 512);  // 125 x 12
  cproj_gemm_kernel<<<gridA, dim3(256), 0, stream>>>(c, Wk, cproj);

  const int nwaves = BB * LW * KK;           // 24000
  dim3 gridB((nwaves + 7) / 8);              // 3000 blocks of 8 waves
  dots_kernel<<<gridB, dim3(256), 0, stream>>>(cproj, z, perms_len, perm_L,
                                               perm_B, out);
}